// Assigner_84894323573283
// MI455X (gfx1250) — compile-verified
//
#include <hip/hip_runtime.h>
#include <hip/hip_bf16.h>
#include <math.h>

// ---------------- problem constants (from reference) ----------------
constexpr int B_ = 32;
constexpr int A_ = 8400;
constexpr int M_ = 32;
constexpr int NC_ = 80;
constexpr int TOPK_ = 13;
constexpr float EPS_ = 1e-9f;
constexpr float CEPS_ = 1e-7f;
constexpr float FOUR_OVER_PI2 = 0.4052847345693511f; // 4/pi^2

typedef __attribute__((ext_vector_type(2))) float v2f;
typedef __attribute__((ext_vector_type(8))) float v8f;

// =====================================================================
// K1: gathered[b][m][a] = pd_scores[b][a][label[b][m]]
//     computed as OneHot(labels)[16x4] x pd_scores^T[4x16] via
//     V_WMMA_F32_16X16X4_F32, accumulating K = NC = 80 in 20 steps.
//     One wave32 per 16(M) x 16(A) tile. Exact in f32 (one-hot rows).
// =====================================================================
__global__ __launch_bounds__(32) void k1_gather_wmma(
    const float* __restrict__ pd_scores,   // (B, A, NC)
    const int*   __restrict__ gt_labels,   // (B, M, 1)
    float*       __restrict__ gathered)    // (B, M, A)
{
    const int atile = blockIdx.x;     // 0..524
    const int mtile = blockIdx.y;     // 0..1
    const int b     = blockIdx.z;     // 0..31
    const int lane  = threadIdx.x;    // 0..31 (one wave32)
    const int half  = lane >> 4;      // 0: K={0,1}, 1: K={2,3} within a k-slice
    const int l16   = lane & 15;

    // A-fragment source: one-hot of this lane's GT label (row M = l16)
    const int label = gt_labels[b * M_ + mtile * 16 + l16];

    // B-fragment source: pd_scores row for anchor column N = l16
    const int acol = atile * 16 + l16;
    const float* prow = pd_scores + ((size_t)b * A_ + acol) * NC_ + 2 * half;

    v8f acc = {};
#pragma unroll
    for (int kk = 0; kk < NC_ / 4; ++kk) {
        const int k0 = 4 * kk + 2 * half;
        v2f af;
        af.x = (label == k0    ) ? 1.0f : 0.0f;
        af.y = (label == k0 + 1) ? 1.0f : 0.0f;
        // 8-byte aligned: NC even, offsets even
        v2f bf = *reinterpret_cast<const v2f*>(prow + 4 * kk);
        acc = __builtin_amdgcn_wmma_f32_16x16x4_f32(
            /*neg_a=*/false, af, /*neg_b=*/false, bf,
            /*c_mod=*/(short)0, acc, /*reuse_a=*/false, /*reuse_b=*/false);
    }

    // D layout: VGPR r -> lanes 0-15: M=r, lanes 16-31: M=r+8; N = l16
    float* orow = gathered + ((size_t)b * M_ + mtile * 16 + 8 * half) * A_
                + atile * 16 + l16;
#pragma unroll
    for (int r = 0; r < 8; ++r) orow[(size_t)r * A_] = acc[r];
}

// =====================================================================
// K2: CIoU + masks + align metric.
//     overlaps[b][m][a]   = gt_mask ? max(ciou,0) : 0
//     alignbuf[b][m][a]   = (gt_mask ? gathered : 0) * overlaps^6
//     (alignbuf aliases the gathered buffer; same element, in-place)
// =====================================================================
__global__ __launch_bounds__(256) void k2_metrics(
    const float* __restrict__ pd_bboxes,   // (B, A, 4)
    const float* __restrict__ anc_points,  // (A, 2)
    const float* __restrict__ gt_bboxes,   // (B, M, 4)
    const float* __restrict__ mask_gt,     // (B, M, 1)
    float*       __restrict__ alignbuf,    // (B, M, A) in: gathered, out: align
    float*       __restrict__ overlaps)    // (B, M, A)
{
    const int a = blockIdx.x * 256 + threadIdx.x;
    if (a >= A_) return;
    const int m = blockIdx.y;
    const int b = blockIdx.z;

    const float4 g  = reinterpret_cast<const float4*>(gt_bboxes)[b * M_ + m];
    const float  mg = mask_gt[b * M_ + m];
    const float2 ap = reinterpret_cast<const float2*>(anc_points)[a];
    const float4 p  = reinterpret_cast<const float4*>(pd_bboxes)[(size_t)b * A_ + a];

    const float din = fminf(fminf(ap.x - g.x, ap.y - g.y),
                            fminf(g.z - ap.x, g.w - ap.y));
    const bool in_gts  = din > EPS_;
    const bool gt_mask = in_gts && (mg > 0.0f);

    // ---- CIoU(gt=b1, pd=b2), eps = 1e-7 ----
    const float w1 = g.z - g.x, h1 = g.w - g.y + CEPS_;
    const float w2 = p.z - p.x, h2 = p.w - p.y + CEPS_;
    const float iw = fmaxf(fminf(g.z, p.z) - fmaxf(g.x, p.x), 0.0f);
    const float ih = fmaxf(fminf(g.w, p.w) - fmaxf(g.y, p.y), 0.0f);
    const float inter = iw * ih;
    const float uni   = w1 * h1 + w2 * h2 - inter + CEPS_;
    const float iou   = inter / uni;
    const float cw = fmaxf(g.z, p.z) - fminf(g.x, p.x);
    const float ch = fmaxf(g.w, p.w) - fminf(g.y, p.y);
    const float c2 = cw * cw + ch * ch + CEPS_;
    const float dx = p.x + p.z - g.x - g.z;
    const float dy = p.y + p.w - g.y - g.w;
    const float rho2 = (dx * dx + dy * dy) * 0.25f;
    const float dv = atanf(w2 / h2) - atanf(w1 / h1);
    const float v  = FOUR_OVER_PI2 * dv * dv;
    const float alpha = v / (v - iou + 1.0f + CEPS_);
    const float ciou  = iou - (rho2 / c2 + v * alpha);

    const float ov = gt_mask ? fmaxf(ciou, 0.0f) : 0.0f;
    const size_t idx = ((size_t)b * M_ + m) * A_ + a;
    const float sc = gt_mask ? alignbuf[idx] : 0.0f;   // gathered score
    const float ov2 = ov * ov;
    const float ov6 = ov2 * ov2 * ov2;                  // BETA = 6
    overlaps[idx] = ov;
    alignbuf[idx] = sc * ov6;                           // ALPHA = 1
}

// =====================================================================
// K3: top-13 per (b,m) row over A, lowest-index tie-break (== lax.top_k),
//     then maskPos[b][m][a in top13] = mask_in_gts. mg==0 rows stay zero.
// =====================================================================
__global__ __launch_bounds__(256) void k3_topk(
    const float* __restrict__ alignbuf,    // (B, M, A)
    const float* __restrict__ mask_gt,     // (B, M, 1)
    const float* __restrict__ anc_points,  // (A, 2)
    const float* __restrict__ gt_bboxes,   // (B, M, 4)
    unsigned char* __restrict__ maskPos)   // (B, M, A)
{
    const int m = blockIdx.x;
    const int b = blockIdx.y;
    const int tid = threadIdx.x;
    const size_t base = ((size_t)b * M_ + m) * A_;

    __shared__ float smet[A_];
    __shared__ float sval[256];
    __shared__ int   sidx[256];
    __shared__ int   ssel[TOPK_];

    const bool mg = mask_gt[b * M_ + m] > 0.0f;   // block-uniform

    for (int i = tid; i < A_; i += 256) {
        smet[i] = alignbuf[base + i];
        maskPos[base + i] = 0;
    }
    __syncthreads();

    if (mg) {
        for (int r = 0; r < TOPK_; ++r) {
            float bv = -1e30f;
            int   bi = 0x7fffffff;
            for (int i = tid; i < A_; i += 256) {
                const float v = smet[i];
                if (v > bv || (v == bv && i < bi)) { bv = v; bi = i; }
            }
            sval[tid] = bv; sidx[tid] = bi;
            __syncthreads();
            for (int s = 128; s > 0; s >>= 1) {
                if (tid < s) {
                    const float v = sval[tid + s];
                    const int   ii = sidx[tid + s];
                    if (v > sval[tid] || (v == sval[tid] && ii < sidx[tid])) {
                        sval[tid] = v; sidx[tid] = ii;
                    }
                }
                __syncthreads();
            }
            if (tid == 0) { ssel[r] = sidx[0]; smet[sidx[0]] = -1e38f; }
            __syncthreads();
        }
        if (tid < TOPK_) {
            const int a = ssel[tid];
            const float4 g  = reinterpret_cast<const float4*>(gt_bboxes)[b * M_ + m];
            const float2 ap = reinterpret_cast<const float2*>(anc_points)[a];
            const float d = fminf(fminf(ap.x - g.x, ap.y - g.y),
                                  fminf(g.z - ap.x, g.w - ap.y));
            maskPos[base + a] = (d > EPS_) ? 1 : 0;
        }
    }
}

// =====================================================================
// K4: per (b,a): resolve multi-GT anchors (first-argmax of overlaps),
//     finalize maskPos, emit target_gt_idx, target_bboxes, fg_mask.
// =====================================================================
__global__ __launch_bounds__(256) void k4_resolve(
    const float* __restrict__ overlaps,    // (B, M, A)
    unsigned char* __restrict__ maskPos,   // (B, M, A) updated in place
    const float* __restrict__ gt_bboxes,   // (B, M, 4)
    int*   __restrict__ tgtIdx,            // (B, A)
    float* __restrict__ outBbox,           // (B, A, 4)
    float* __restrict__ outFg)             // (B, A)
{
    const int idx = blockIdx.x * 256 + threadIdx.x;
    if (idx >= B_ * A_) return;
    const int b = idx / A_;
    const int a = idx - b * A_;
    const size_t cbase = (size_t)b * M_ * A_ + a;

    unsigned char mp[M_];
    int fg = 0;
#pragma unroll
    for (int m = 0; m < M_; ++m) { mp[m] = maskPos[cbase + (size_t)m * A_]; fg += mp[m]; }

    if (fg > 1) {
        float bo = -1e30f; int bm = 0;
#pragma unroll
        for (int m = 0; m < M_; ++m) {
            const float o = overlaps[cbase + (size_t)m * A_];
            if (o > bo) { bo = o; bm = m; }            // first max wins ties
        }
#pragma unroll
        for (int m = 0; m < M_; ++m) {
            const unsigned char nv = (m == bm) ? 1 : 0;
            maskPos[cbase + (size_t)m * A_] = nv;
            mp[m] = nv;
        }
        fg = 1;
    }

    int tgt = 0;
#pragma unroll
    for (int m = M_ - 1; m >= 0; --m) { if (mp[m]) tgt = m; }  // first set bit, else 0

    tgtIdx[idx] = tgt;
    reinterpret_cast<float4*>(outBbox)[idx] =
        reinterpret_cast<const float4*>(gt_bboxes)[b * M_ + tgt];
    outFg[idx] = (fg > 0) ? 1.0f : 0.0f;
}

// =====================================================================
// K5: per (b,m): pos_am = max_a align*mask ; pos_ov = max_a overlaps*mask
// =====================================================================
__global__ __launch_bounds__(256) void k5_rowmax(
    const float* __restrict__ alignbuf,
    const float* __restrict__ overlaps,
    const unsigned char* __restrict__ maskPos,
    float* __restrict__ posAm, float* __restrict__ posOv)
{
    const int m = blockIdx.x;
    const int b = blockIdx.y;
    const size_t base = ((size_t)b * M_ + m) * A_;
    __shared__ float sa[256], so[256];

    float ma = 0.0f, mo = 0.0f;   // all products are >= 0
    for (int i = threadIdx.x; i < A_; i += 256) {
        const float mk = maskPos[base + i] ? 1.0f : 0.0f;
        ma = fmaxf(ma, alignbuf[base + i] * mk);
        mo = fmaxf(mo, overlaps[base + i] * mk);
    }
    sa[threadIdx.x] = ma; so[threadIdx.x] = mo;
    __syncthreads();
    for (int s = 128; s > 0; s >>= 1) {
        if (threadIdx.x < s) {
            sa[threadIdx.x] = fmaxf(sa[threadIdx.x], sa[threadIdx.x + s]);
            so[threadIdx.x] = fmaxf(so[threadIdx.x], so[threadIdx.x + s]);
        }
        __syncthreads();
    }
    if (threadIdx.x == 0) { posAm[b * M_ + m] = sa[0]; posOv[b * M_ + m] = so[0]; }
}

// =====================================================================
// K6: per (b,a): norm = max_m (align*mask * pos_ov / (pos_am + EPS));
//     target label = max(labels[b, tgt], 0).
// =====================================================================
__global__ __launch_bounds__(256) void k6_norm(
    const float* __restrict__ alignbuf,
    const unsigned char* __restrict__ maskPos,
    const float* __restrict__ posAm, const float* __restrict__ posOv,
    const int* __restrict__ gt_labels, const int* __restrict__ tgtIdx,
    float* __restrict__ normBuf, int* __restrict__ tlBuf)
{
    const int idx = blockIdx.x * 256 + threadIdx.x;
    if (idx >= B_ * A_) return;
    const int b = idx / A_;
    const int a = idx - b * A_;
    const size_t cbase = (size_t)b * M_ * A_ + a;

    float norm = 0.0f;
#pragma unroll
    for (int m = 0; m < M_; ++m) {
        if (maskPos[cbase + (size_t)m * A_]) {
            const float t = alignbuf[cbase + (size_t)m * A_]
                          * posOv[b * M_ + m] / (posAm[b * M_ + m] + EPS_);
            norm = fmaxf(norm, t);
        }
    }
    normBuf[idx] = norm;
    const int tgt = tgtIdx[idx];
    int tl = gt_labels[b * M_ + tgt];
    tlBuf[idx] = (tl > 0) ? tl : 0;
}

// =====================================================================
// K7: coalesced one-hot score write: (B, A, NC)
// =====================================================================
__global__ __launch_bounds__(256) void k7_scores(
    const float* __restrict__ normBuf, const int* __restrict__ tlBuf,
    const float* __restrict__ outFg, float* __restrict__ outScores)
{
    const long long i = (long long)blockIdx.x * 256 + threadIdx.x;
    if (i >= (long long)B_ * A_ * NC_) return;
    const int c  = (int)(i % NC_);
    const long long ba = i / NC_;
    const float v = (outFg[ba] > 0.0f && c == tlBuf[ba]) ? normBuf[ba] : 0.0f;
    outScores[i] = v;
}

// =====================================================================
extern "C" void kernel_launch(void* const* d_in, const int* in_sizes, int n_in,
                              void* d_out, int out_size, void* d_ws, size_t ws_size,
                              hipStream_t stream) {
    const float* pd_scores  = (const float*)d_in[0];   // (B, A, NC)
    const float* pd_bboxes  = (const float*)d_in[1];   // (B, A, 4)
    const float* anc_points = (const float*)d_in[2];   // (A, 2)
    const int*   gt_labels  = (const int*)  d_in[3];   // (B, M, 1)
    const float* gt_bboxes  = (const float*)d_in[4];   // (B, M, 4)
    const float* mask_gt    = (const float*)d_in[5];   // (B, M, 1)

    float* outBbox   = (float*)d_out;                        // (B, A, 4)
    float* outScores = outBbox + (size_t)B_ * A_ * 4;        // (B, A, NC)
    float* outFg     = outScores + (size_t)B_ * A_ * NC_;    // (B, A)

    // ---- workspace carve-up (256B aligned slices) ----
    const size_t nBMA = (size_t)B_ * M_ * A_;
    auto align256 = [](size_t x) { return (x + 255) & ~(size_t)255; };
    char* ws = (char*)d_ws;
    size_t off = 0;
    float* alignbuf = (float*)(ws + off); off += align256(nBMA * sizeof(float));
    float* overlaps = (float*)(ws + off); off += align256(nBMA * sizeof(float));
    unsigned char* maskPos = (unsigned char*)(ws + off); off += align256(nBMA);
    int*   tgtIdx  = (int*)  (ws + off); off += align256((size_t)B_ * A_ * sizeof(int));
    float* posAm   = (float*)(ws + off); off += align256((size_t)B_ * M_ * sizeof(float));
    float* posOv   = (float*)(ws + off); off += align256((size_t)B_ * M_ * sizeof(float));
    float* normBuf = (float*)(ws + off); off += align256((size_t)B_ * A_ * sizeof(float));
    int*   tlBuf   = (int*)  (ws + off); off += align256((size_t)B_ * A_ * sizeof(int));
    (void)ws_size; (void)in_sizes; (void)n_in; (void)out_size;

    // K1: WMMA one-hot gather (one wave32 per 16x16 tile)
    k1_gather_wmma<<<dim3(A_ / 16, M_ / 16, B_), 32, 0, stream>>>(
        pd_scores, gt_labels, alignbuf);

    // K2: CIoU + align metric
    k2_metrics<<<dim3((A_ + 255) / 256, M_, B_), 256, 0, stream>>>(
        pd_bboxes, anc_points, gt_bboxes, mask_gt, alignbuf, overlaps);

    // K3: top-13 per (b,m)
    k3_topk<<<dim3(M_, B_), 256, 0, stream>>>(
        alignbuf, mask_gt, anc_points, gt_bboxes, maskPos);

    // K4: resolve multi-assignment, bboxes, fg
    k4_resolve<<<(B_ * A_ + 255) / 256, 256, 0, stream>>>(
        overlaps, maskPos, gt_bboxes, tgtIdx, outBbox, outFg);

    // K5: per-(b,m) row maxes
    k5_rowmax<<<dim3(M_, B_), 256, 0, stream>>>(
        alignbuf, overlaps, maskPos, posAm, posOv);

    // K6: per-(b,a) norm + target label
    k6_norm<<<(B_ * A_ + 255) / 256, 256, 0, stream>>>(
        alignbuf, maskPos, posAm, posOv, gt_labels, tgtIdx, normBuf, tlBuf);

    // K7: one-hot score write
    const long long nScore = (long long)B_ * A_ * NC_;
    k7_scores<<<(unsigned)((nScore + 255) / 256), 256, 0, stream>>>(
        normBuf, tlBuf, outFg, outScores);
}